// VICRegLLoss_51316269253225
// MI455X (gfx1250) — compile-verified
//
#include <hip/hip_runtime.h>
#include <hip/hip_bf16.h>
#include <math.h>

// ---------------------------------------------------------------------------
// VICRegL loss for MI455X (gfx1250, wave32).
// Heavy math (feature cdist GEMM, 32x32 Gram) runs on V_WMMA_F32_16X16X4_F32.
// cdist is LDS-tiled; tiles staged with gfx1250 async global->LDS copies.
// ---------------------------------------------------------------------------

typedef __attribute__((ext_vector_type(2))) float v2f;
typedef __attribute__((ext_vector_type(8))) float v8f;
typedef int v4i_t __attribute__((vector_size(16)));   // b128 transfer unit

#define NB    32      // batch
#define DG    8192    // global embedding dim
#define LG    576     // 24*24 global locations
#define LL    256     // 16*16 local locations
#define CF    512     // local feature channels
#define MG    20      // num matches, global side
#define ML    4       // num matches, local side
#define KC    64      // K-chunk staged in LDS per iteration

#if defined(__has_builtin)
#if __has_builtin(__builtin_amdgcn_global_load_async_to_lds_b128)
#define HAVE_ASYNC_LDS 1
#endif
#endif

#ifdef HAVE_ASYNC_LDS
// prototype: (int4 __device__* gsrc, int4 __shared__* ldst, imm offset, imm cpol)
#define ASYNC_COPY16(gp, lp)                                         \
  __builtin_amdgcn_global_load_async_to_lds_b128(                    \
      (v4i_t*)(gp), (v4i_t*)(lp), 0, 0)
__device__ __forceinline__ void wait_async_all() {
#if __has_builtin(__builtin_amdgcn_s_wait_asynccnt)
  __builtin_amdgcn_s_wait_asynccnt(0);
#else
  asm volatile("s_wait_asynccnt 0" ::: "memory");
#endif
}
#else
#define ASYNC_COPY16(gp, lp)                                         \
  do { float4 _v = *(const float4*)(gp); *(float4*)(lp) = _v; } while (0)
__device__ __forceinline__ void wait_async_all() {}
#endif

__device__ __forceinline__ float warp_sum32(float v) {
#pragma unroll
  for (int m = 16; m > 0; m >>= 1) v += __shfl_xor(v, m, 32);
  return v;
}

__device__ __forceinline__ v8f wmma_f32x4(v2f a, v2f b, v8f c) {
  return __builtin_amdgcn_wmma_f32_16x16x4_f32(false, a, false, b,
                                               (short)0, c, false, false);
}

// ---------------------------------------------------------------------------
// 0) zero scalar accumulators + Gram accumulators
// acc layout: 0 inv_sum, 1 vsumA, 2 vsumB, 3 dsumA, 4 dsumB, 5..8 mse sums
// ---------------------------------------------------------------------------
__global__ void vrl_init(float* __restrict__ acc, float* __restrict__ Ga,
                         float* __restrict__ Gb) {
  int t = threadIdx.x;
  for (int i = t; i < 16; i += 256) acc[i] = 0.0f;
  for (int i = t; i < 1024; i += 256) { Ga[i] = 0.0f; Gb[i] = 0.0f; }
}

// ---------------------------------------------------------------------------
// 1) per-row squared norms of the feature maps (one wave32 per row)
// ---------------------------------------------------------------------------
__global__ void vrl_row_norms(const float* __restrict__ fg,
                              const float* __restrict__ fl,
                              float* __restrict__ xn, float* __restrict__ yn) {
  int wid  = (int)((blockIdx.x * blockDim.x + threadIdx.x) >> 5);
  int lane = threadIdx.x & 31;
  const int NX = NB * LG;
  const float* src;
  float* dst;
  if (wid < NX) {
    src = fg + (size_t)wid * CF; dst = xn + wid;
  } else {
    int r = wid - NX;
    if (r >= NB * LL) return;
    src = fl + (size_t)r * CF; dst = yn + r;
  }
  float s = 0.0f;
  for (int c = lane; c < CF; c += 32) { float v = src[c]; s += v * v; }
  s = warp_sum32(s);
  if (lane == 0) *dst = s;
}

// ---------------------------------------------------------------------------
// 2) feature-space cdist D[b,i,n] = ||zg[b,i] - zl[b,n]|| via f32 WMMA.
//    256-thread workgroup computes a 64x128 output block of one batch.
//    K staged in LDS chunks of 64 (async global->LDS, ASYNCcnt-tracked);
//    row stride padded to 68 floats -> conflict-free ds_load_b64 fragments.
//    8 waves: wave w -> m-tile (w>>1), n-half (w&1) of 4 consecutive 16x16
//    WMMA tiles (4 accumulators per wave).
//    A (16x4, MxK): lane<16 -> K=k0+{0,1}; lane>=16 -> K=k0+{2,3}
//    B (4x16, KxN): mirrored layout, sourced row-major (Y rows = B columns).
// ---------------------------------------------------------------------------
__global__ __launch_bounds__(256) void vrl_cdist_wmma(
    const float* __restrict__ fg, const float* __restrict__ fl,
    const float* __restrict__ xn, const float* __restrict__ yn,
    float* __restrict__ Dm) {
  const int b    = blockIdx.y;
  const int bm   = blockIdx.x >> 1;          // 0..8  : 64-row strip
  const int bn   = blockIdx.x & 1;           // 0..1  : 128-col strip
  const int tid  = threadIdx.x;
  const int lane = tid & 31;
  const int w    = tid >> 5;                 // wave 0..7
  const int wm   = w >> 1;                   // 0..3 : m-tile within strip
  const int wn   = w & 1;                    // 0..1 : 64-col half
  const int half = lane >> 4;
  const int r    = lane & 15;

  __shared__ __attribute__((aligned(16))) float Xs[64][68];
  __shared__ __attribute__((aligned(16))) float Ys[128][68];

  const int m0 = bm * 64;
  const int n0 = bn * 128;
  const float* Xg = fg + ((size_t)b * LG + m0) * CF;
  const float* Yg = fl + ((size_t)b * LL + n0) * CF;

  v8f acc[4] = {v8f{}, v8f{}, v8f{}, v8f{}};

  for (int ch = 0; ch < CF / KC; ++ch) {
    // ---- stage A strip: 64 rows x 64 floats = 1024 x b128 transfers
#pragma unroll
    for (int rep = 0; rep < 4; ++rep) {
      int idx = rep * 256 + tid;
      int row = idx >> 4, q = idx & 15;
      const float* gp = Xg + (size_t)row * CF + ch * KC + q * 4;
      float* lp = &Xs[row][q * 4];
      ASYNC_COPY16(gp, lp);
    }
    // ---- stage B strip: 128 rows x 64 floats = 2048 x b128 transfers
#pragma unroll
    for (int rep = 0; rep < 8; ++rep) {
      int idx = rep * 256 + tid;
      int row = idx >> 4, q = idx & 15;
      const float* gp = Yg + (size_t)row * CF + ch * KC + q * 4;
      float* lp = &Ys[row][q * 4];
      ASYNC_COPY16(gp, lp);
    }
    wait_async_all();
    __syncthreads();

    // ---- consume chunk: 16 K-steps x 4 WMMA tiles
#pragma unroll
    for (int kk = 0; kk < KC / 4; ++kk) {
      const int col = kk * 4 + half * 2;
      v2f a = *(const v2f*)&Xs[wm * 16 + r][col];
      v2f b0 = *(const v2f*)&Ys[wn * 64 + 0 + r][col];
      v2f b1 = *(const v2f*)&Ys[wn * 64 + 16 + r][col];
      v2f b2 = *(const v2f*)&Ys[wn * 64 + 32 + r][col];
      v2f b3 = *(const v2f*)&Ys[wn * 64 + 48 + r][col];
      acc[0] = wmma_f32x4(a, b0, acc[0]);
      acc[1] = wmma_f32x4(a, b1, acc[1]);
      acc[2] = wmma_f32x4(a, b2, acc[2]);
      acc[3] = wmma_f32x4(a, b3, acc[3]);
    }
    __syncthreads();
  }

  // ---- epilogue: d = sqrt(max(x2 + y2 - 2*dot, 0))
  const int gm_base = m0 + wm * 16;
#pragma unroll
  for (int nt = 0; nt < 4; ++nt) {
    const int n  = n0 + wn * 64 + nt * 16 + r;
    const float y2 = yn[b * LL + n];
#pragma unroll
    for (int v = 0; v < 8; ++v) {
      const int m = gm_base + v + half * 8;
      float d2 = xn[b * LG + m] + y2 - 2.0f * acc[nt][v];
      Dm[((size_t)b * LG + m) * LL + n] = sqrtf(fmaxf(d2, 0.0f));
    }
  }
}

// ---------------------------------------------------------------------------
// 3) 32x32 Gram matrices G = Z Z^T (K = 8192) via f32 WMMA, K split 16 ways.
// ---------------------------------------------------------------------------
__global__ void vrl_gram_wmma(const float* __restrict__ za,
                              const float* __restrict__ zb,
                              float* __restrict__ Ga, float* __restrict__ Gb) {
  const int g  = blockIdx.x;            // 0..7: {matrix, 2x2 tile}
  const int kc = blockIdx.y * (DG / 16);
  const float* Z = (g < 4) ? za : zb;
  float* G       = (g < 4) ? Ga : Gb;
  const int tile = g & 3;
  const int tm = tile >> 1, tn = tile & 1;
  const int lane = threadIdx.x, half = lane >> 4, r = lane & 15;

  const float2* X = (const float2*)(Z + (size_t)(tm * 16 + r) * DG + kc);
  const float2* Y = (const float2*)(Z + (size_t)(tn * 16 + r) * DG + kc);

  v8f acc = {};
  for (int it = 0; it < (DG / 16) / 4; ++it) {
    int h = it * 2 + half;
    float2 ax = X[h];
    float2 by = Y[h];
    v2f a  = {ax.x, ax.y};
    v2f bb = {by.x, by.y};
    acc = wmma_f32x4(a, bb, acc);
  }
  const int n = tn * 16 + r;
#pragma unroll
  for (int v = 0; v < 8; ++v) {
    int m = tm * 16 + v + half * 8;
    atomicAdd(&G[m * 32 + n], acc[v]);
  }
}

// ---------------------------------------------------------------------------
// 4) per-feature column stats of z_global/z_local (n=32, ddof=1) and the
//    invariance MSE sum, reduced into scalar accumulators.
// ---------------------------------------------------------------------------
__global__ void vrl_colstats(const float* __restrict__ za,
                             const float* __restrict__ zb,
                             float* __restrict__ acc) {
  const int j = blockIdx.x * blockDim.x + threadIdx.x;   // feature index
  float sa = 0.f, s2a = 0.f, sb = 0.f, s2b = 0.f, invp = 0.f;
#pragma unroll 4
  for (int i = 0; i < NB; ++i) {
    float va = za[(size_t)i * DG + j];
    float vb = zb[(size_t)i * DG + j];
    sa += va; s2a += va * va;
    sb += vb; s2b += vb * vb;
    float d = va - vb; invp += d * d;
  }
  float mua  = sa * (1.0f / NB);
  float vara = (s2a - NB * mua * mua) * (1.0f / (NB - 1));
  float stda = sqrtf(vara + 1e-4f);
  float mub  = sb * (1.0f / NB);
  float varb = (s2b - NB * mub * mub) * (1.0f / (NB - 1));
  float stdb = sqrtf(varb + 1e-4f);

  float ra = fmaxf(1.0f - stda, 0.0f), rb = fmaxf(1.0f - stdb, 0.0f);
  float da = vara * vara,              db = varb * varb;

  invp = warp_sum32(invp);
  ra = warp_sum32(ra); rb = warp_sum32(rb);
  da = warp_sum32(da); db = warp_sum32(db);
  if ((threadIdx.x & 31) == 0) {
    atomicAdd(&acc[0], invp);
    atomicAdd(&acc[1], ra);
    atomicAdd(&acc[2], rb);
    atomicAdd(&acc[3], da);
    atomicAdd(&acc[4], db);
  }
}

// ---------------------------------------------------------------------------
// 5) nearest-neighbor selection + gather + MSE partial sums.
//    case 0: feat, global->local (M=20, dist = D rows)
//    case 1: feat, local->global (M=4,  dist = D cols)
//    case 2: grid, global->local (M=20, dist from 2-D grids, gather features)
//    case 3: grid, local->global (M=4)
//    Selection = rank < M with ties broken toward lower index (matches
//    jax.lax.top_k); order is irrelevant because only the mean is taken.
// ---------------------------------------------------------------------------
__global__ void vrl_nn_mse(const float* __restrict__ fg,
                           const float* __restrict__ fl,
                           const float* __restrict__ gg,
                           const float* __restrict__ gl,
                           const float* __restrict__ Dm,
                           float* __restrict__ acc) {
  const int cs  = blockIdx.x;
  const int b   = blockIdx.y;
  const int tid = threadIdx.x;
  const bool gside = (cs == 0) || (cs == 2);   // input = global map
  const int L1 = gside ? LG : LL;
  const int L2 = gside ? LL : LG;
  const int M  = gside ? MG : ML;
  const float* inF = gside ? fg + (size_t)b * LG * CF : fl + (size_t)b * LL * CF;
  const float* caF = gside ? fl + (size_t)b * LL * CF : fg + (size_t)b * LG * CF;

  __shared__ float nnd[LG];
  __shared__ int   nni[LG];
  __shared__ float cg[LG * 2];
  __shared__ int   pairsI[MG], pairsN[MG];
  __shared__ int   cnt;
  __shared__ float red[256];

  if (tid == 0) cnt = 0;
  if (cs >= 2) {
    const float* cand_grid =
        (cs == 2) ? gl + (size_t)b * LL * 2 : gg + (size_t)b * LG * 2;
    for (int t = tid; t < L2 * 2; t += 256) cg[t] = cand_grid[t];
  }
  __syncthreads();

  const float* gIn =
      (cs == 2) ? gg + (size_t)b * LG * 2 :
      (cs == 3) ? gl + (size_t)b * LL * 2 : (const float*)0;

  // nearest candidate per input row
  for (int i = tid; i < L1; i += 256) {
    float best = 3.402823e38f;
    int   bi   = 0;
    if (cs == 0) {
      const float* row = Dm + ((size_t)b * LG + i) * LL;
      for (int n = 0; n < L2; ++n) {
        float v = row[n];
        if (v < best) { best = v; bi = n; }
      }
    } else if (cs == 1) {
      const float* col = Dm + (size_t)b * LG * LL + i;
      for (int n = 0; n < L2; ++n) {
        float v = col[(size_t)n * LL];
        if (v < best) { best = v; bi = n; }
      }
    } else {
      float xi = gIn[i * 2], yi = gIn[i * 2 + 1];
      for (int n = 0; n < L2; ++n) {       // squared dist: argmin/rank-safe
        float dx = xi - cg[n * 2], dy = yi - cg[n * 2 + 1];
        float v = dx * dx + dy * dy;
        if (v < best) { best = v; bi = n; }
      }
    }
    nnd[i] = best;
    nni[i] = bi;
  }
  __syncthreads();

  // rank selection: exactly M winners (total order via index tie-break)
  for (int i = tid; i < L1; i += 256) {
    float di = nnd[i];
    int rank = 0;
    for (int j = 0; j < L1; ++j) {
      float dj = nnd[j];
      rank += (dj < di) || (dj == di && j < i);
    }
    if (rank < M) {
      int p = atomicAdd(&cnt, 1);
      pairsI[p] = i;
      pairsN[p] = nni[i];
    }
  }
  __syncthreads();

  // gather + squared-error partial sum over M*CF elements
  float local = 0.0f;
  for (int t = tid; t < M * CF; t += 256) {
    int m = t >> 9, c = t & (CF - 1);
    int i = pairsI[m], n = pairsN[m];
    float d = inF[(size_t)i * CF + c] - caF[(size_t)n * CF + c];
    local += d * d;
  }
  red[tid] = local;
  __syncthreads();
  for (int s = 128; s > 0; s >>= 1) {
    if (tid < s) red[tid] += red[tid + s];
    __syncthreads();
  }
  if (tid == 0) atomicAdd(&acc[5 + cs], red[0]);
}

// ---------------------------------------------------------------------------
// 6) finalize: double-center Gram (Gc = HGH), assemble scalar loss.
//    sum(cov^2) = ||Gc||_F^2 / (n-1)^2 ; sum(diag^2) = sum var_j^2.
// ---------------------------------------------------------------------------
__global__ void vrl_finalize(const float* __restrict__ Ga,
                             const float* __restrict__ Gb,
                             const float* __restrict__ acc,
                             float* __restrict__ out) {
  __shared__ float rmA[32], rmB[32], red[256];
  __shared__ float totA, totB, faS;
  const int tid = threadIdx.x;

  if (tid < 32) {
    float s = 0.f;
    for (int k = 0; k < 32; ++k) s += Ga[tid * 32 + k];
    rmA[tid] = s * (1.0f / 32.0f);
  } else if (tid < 64) {
    int i = tid - 32;
    float s = 0.f;
    for (int k = 0; k < 32; ++k) s += Gb[i * 32 + k];
    rmB[i] = s * (1.0f / 32.0f);
  }
  __syncthreads();
  if (tid == 0) {
    float s = 0.f, t = 0.f;
    for (int i = 0; i < 32; ++i) { s += rmA[i]; t += rmB[i]; }
    totA = s * (1.0f / 32.0f);
    totB = t * (1.0f / 32.0f);
  }
  __syncthreads();

  float la = 0.f, lb = 0.f;
  for (int t = tid; t < 1024; t += 256) {
    int i = t >> 5, k = t & 31;
    float ga = Ga[t] - rmA[i] - rmA[k] + totA;
    float gb = Gb[t] - rmB[i] - rmB[k] + totB;
    la += ga * ga;
    lb += gb * gb;
  }
  red[tid] = la;
  __syncthreads();
  for (int s = 128; s > 0; s >>= 1) { if (tid < s) red[tid] += red[tid + s]; __syncthreads(); }
  if (tid == 0) faS = red[0];
  __syncthreads();
  red[tid] = lb;
  __syncthreads();
  for (int s = 128; s > 0; s >>= 1) { if (tid < s) red[tid] += red[tid + s]; __syncthreads(); }

  if (tid == 0) {
    const float fa = faS, fb = red[0];
    const float inv  = acc[0] * (1.0f / (NB * (float)DG));
    const float va   = acc[1] * (1.0f / DG);
    const float vb   = acc[2] * (1.0f / DG);
    const float cova = (fa * (1.0f / 961.0f) - acc[3]) * (1.0f / DG);
    const float covb = (fb * (1.0f / 961.0f) - acc[4]) * (1.0f / DG);
    const float global_loss = 25.0f * inv + 25.0f * 0.5f * (va + vb)
                              + 1.0f * (cova + covb);
    const float mse0 = acc[5] * (1.0f / (NB * MG * (float)CF));
    const float mse1 = acc[6] * (1.0f / (NB * ML * (float)CF));
    const float mse2 = acc[7] * (1.0f / (NB * MG * (float)CF));
    const float mse3 = acc[8] * (1.0f / (NB * ML * (float)CF));
    const float local_loss = 25.0f * 0.5f * (mse0 + mse1 + mse2 + mse3);
    out[0] = 0.25f * global_loss + 0.75f * local_loss;
  }
}

// ---------------------------------------------------------------------------
extern "C" void kernel_launch(void* const* d_in, const int* in_sizes, int n_in,
                              void* d_out, int out_size, void* d_ws,
                              size_t ws_size, hipStream_t stream) {
  (void)in_sizes; (void)n_in; (void)out_size; (void)ws_size;
  const float* za = (const float*)d_in[0];   // z_global       (32, 8192)
  const float* zb = (const float*)d_in[1];   // z_local        (32, 8192)
  const float* fg = (const float*)d_in[2];   // global feats   (32, 576, 512)
  const float* fl = (const float*)d_in[3];   // local feats    (32, 256, 512)
  const float* gg = (const float*)d_in[4];   // grid_global    (32, 576, 2)
  const float* gl = (const float*)d_in[5];   // grid_local     (32, 256, 2)
  float* out = (float*)d_out;

  float* ws  = (float*)d_ws;
  float* Dm  = ws;                                   // 32*576*256
  float* xn  = Dm + (size_t)NB * LG * LL;            // 32*576
  float* yn  = xn + NB * LG;                         // 32*256
  float* Ga  = yn + NB * LL;                         // 1024
  float* Gb  = Ga + 1024;                            // 1024
  float* acc = Gb + 1024;                            // 16 scalars

  vrl_init<<<1, 256, 0, stream>>>(acc, Ga, Gb);

  {
    const int waves = NB * LG + NB * LL;             // one wave32 per row
    const int blocks = (waves * 32 + 255) / 256;
    vrl_row_norms<<<blocks, 256, 0, stream>>>(fg, fl, xn, yn);
  }

  // 64x128 output block per workgroup: 9 m-strips x 2 n-strips per batch
  vrl_cdist_wmma<<<dim3(18, NB), 256, 0, stream>>>(fg, fl, xn, yn, Dm);

  vrl_gram_wmma<<<dim3(8, 16), 32, 0, stream>>>(za, zb, Ga, Gb);

  vrl_colstats<<<DG / 256, 256, 0, stream>>>(za, zb, acc);

  vrl_nn_mse<<<dim3(4, NB), 256, 0, stream>>>(fg, fl, gg, gl, Dm, acc);

  vrl_finalize<<<1, 256, 0, stream>>>(Ga, Gb, acc, out);
}